// Probe_46231027974423
// MI455X (gfx1250) — compile-verified
//
#include <hip/hip_runtime.h>

// Problem constants (match setup_inputs: B=4096, T=8192, C=8)
#define B_DIM 4096
#define T_DIM 8192
#define C_DIM 8
#define NTH   256
#define PT    (T_DIM / NTH)   // 32 elements per thread

typedef __attribute__((ext_vector_type(2))) float v2f;
typedef __attribute__((ext_vector_type(8))) float v8f;

// Map float bits to order-monotonic unsigned keys and back.
__device__ __forceinline__ unsigned fkey(float f) {
    unsigned u = __float_as_uint(f);
    return u ^ ((u & 0x80000000u) ? 0xFFFFFFFFu : 0x80000000u);
}
__device__ __forceinline__ float finv(unsigned k) {
    unsigned u = (k & 0x80000000u) ? (k ^ 0x80000000u) : ~k;
    return __uint_as_float(u);
}

// One block per (b, c) series of length T. Stages the series into LDS with
// CDNA5 async global->LDS copies (ASYNCcnt), accumulates sum/sumsq/cnt while
// converting to monotonic keys in place, then does two 4-pass radix-256
// selects (ranks 7371, 7372) with wave32 ballot-aggregated histogramming.
__global__ __launch_bounds__(NTH) void stats_kernel(const float* __restrict__ x,
                                                    float* __restrict__ feats) {
    __shared__ unsigned skey[T_DIM];      // 32 KB: raw bits, then keys in place
    __shared__ unsigned hist[256];
    __shared__ unsigned bcast[2];
    __shared__ float rsum[NTH], rsq[NTH], rcnt[NTH];

    const int bid  = blockIdx.x;
    const int b    = bid >> 3;            // C_DIM == 8
    const int c    = bid & 7;
    const int tid  = threadIdx.x;
    const int lane = tid & 31;            // wave32

    // Uniform SGPR base for GVS-mode async loads; per-lane 32-bit byte offset.
    const unsigned long long base =
        (unsigned long long)(x + ((size_t)b * T_DIM) * C_DIM + c);

    // Async global -> LDS staging: per-lane gather (stride 32 B) straight
    // into LDS, no VGPR round-trip. LDS dest address = low 32 bits of the
    // flat shared-memory address.
#pragma unroll
    for (int i = 0; i < PT; ++i) {
        const int t = tid + i * NTH;
        const unsigned goff   = (unsigned)t * (unsigned)(C_DIM * 4);
        const unsigned ldsOff = (unsigned)(unsigned long long)(&skey[t]);
        asm volatile("global_load_async_to_lds_b32 %0, %1, %2"
                     :: "v"(ldsOff), "v"(goff), "s"(base)
                     : "memory");
    }
    asm volatile("s_wait_asynccnt 0x0" ::: "memory");
    __syncthreads();

    // Accumulate running stats and convert raw bits -> radix keys in place.
    float sum = 0.0f, sq = 0.0f;
    int cnt = 0;
#pragma unroll
    for (int i = 0; i < PT; ++i) {
        const int t = tid + i * NTH;      // each thread owns its slots
        const float v = __uint_as_float(skey[t]);
        sum += v;
        sq  += v * v;
        cnt += (v > 0.0f) ? 1 : 0;
        skey[t] = fkey(v);
    }
    rsum[tid] = sum; rsq[tid] = sq; rcnt[tid] = (float)cnt;
    __syncthreads();
    for (int s = NTH / 2; s > 0; s >>= 1) {
        if (tid < s) {
            rsum[tid] += rsum[tid + s];
            rsq[tid]  += rsq[tid + s];
            rcnt[tid] += rcnt[tid + s];
        }
        __syncthreads();
    }

    // quantile position = 0.9*(T-1) = 7371.9 -> ranks 7371 & 7372, frac 0.9
    float sel[2];
    const int ranks[2] = {7371, 7372};
    for (int s = 0; s < 2; ++s) {
        unsigned prefix = 0;
        int rank = ranks[s];
        for (int pass = 0; pass < 4; ++pass) {
            const int shift = 24 - pass * 8;
            const unsigned pmask = (pass == 0) ? 0u : (0xFFFFFFFFu << (shift + 8));
            hist[tid] = 0u;               // NTH == 256 bins
            __syncthreads();
            for (int i = 0; i < PT; ++i) {
                const unsigned u = skey[tid + i * NTH];
                const unsigned d = (u >> shift) & 0xFFu;
                const bool act = ((u & pmask) == prefix);
                // wave-aggregated histogram: one atomic per distinct digit/wave
                unsigned peers = (unsigned)__ballot(act);
#pragma unroll
                for (int j = 0; j < 8; ++j) {
                    const unsigned bj = (unsigned)__ballot((d >> j) & 1u);
                    peers &= ((d >> j) & 1u) ? bj : ~bj;
                }
                if (act) {
                    const unsigned lt = (1u << lane) - 1u;
                    if ((peers & lt) == 0u)   // leader lane of this digit-group
                        atomicAdd(&hist[d], (unsigned)__popc(peers));
                }
            }
            __syncthreads();
            if (tid == 0) {
                unsigned cum = 0; int bin = 0;
                for (; bin < 255; ++bin) {
                    const unsigned h = hist[bin];
                    if (cum + h > (unsigned)rank) break;
                    cum += h;
                }
                bcast[0] = (unsigned)bin;
                bcast[1] = cum;
            }
            __syncthreads();
            prefix |= bcast[0] << shift;
            rank   -= (int)bcast[1];
            __syncthreads();
        }
        sel[s] = finv(prefix);
    }

    if (tid == 0) {
        const float n    = (float)T_DIM;
        const float tot  = rsum[0];
        const float tsq  = rsq[0];
        const float mean = tot / n;
        const float var  = (tsq - tot * mean) / (n - 1.0f);
        const float stdv = sqrtf(fmaxf(var, 0.0f));   // nan_to_num via clamp
        const float q    = 0.1f * sel[0] + 0.9f * sel[1];
        float* f = feats + ((size_t)b * (C_DIM * 4) + c * 4);
        f[0] = q; f[1] = mean; f[2] = rcnt[0]; f[3] = stdv;
    }
}

// feats [4096,32] f32 x W [32] + bias -> out [4096] using V_WMMA_F32_16X16X4_F32.
// One wave per 16-row tile; K=32 in 8 steps of 4. B is replicated across N so
// only the K-layout matters (VGPR0 <-> K{0,2}, VGPR1 <-> K{1,3} by lane half);
// every output column holds the dot product, lanes 0/16 write column 0.
__global__ __launch_bounds__(32) void gemv_wmma(const float* __restrict__ feats,
                                                const float* __restrict__ W,
                                                const float* __restrict__ bias,
                                                float* __restrict__ out) {
    const int lane = threadIdx.x;
    const int m    = lane & 15;
    const int kh   = (lane >> 4) << 1;           // 0 (lanes 0-15) or 2 (lanes 16-31)
    const size_t rowBase = (size_t)blockIdx.x * 16;

    v8f acc = {};
#pragma unroll
    for (int k0 = 0; k0 < 32; k0 += 4) {
        v2f a, bb;
        const float* arow = feats + (rowBase + m) * 32 + k0 + kh;
        a.x  = arow[0];                           // A[m][k0+kh]
        a.y  = arow[1];                           // A[m][k0+kh+1]
        bb.x = W[k0 + kh];                        // B[k0+kh][*]   (replicated over N)
        bb.y = W[k0 + kh + 1];                    // B[k0+kh+1][*]
        acc = __builtin_amdgcn_wmma_f32_16x16x4_f32(
            /*neg_a=*/false, a, /*neg_b=*/false, bb,
            /*c_mod=*/(short)0, acc, /*reuse_a=*/false, /*reuse_b=*/false);
    }
    if (m == 0) {
        const float bv = bias[0];
        const size_t r0 = rowBase + ((size_t)(lane >> 4) << 3);  // rows 0-7 / 8-15
#pragma unroll
        for (int r = 0; r < 8; ++r) out[r0 + r] = acc[r] + bv;
    }
}

extern "C" void kernel_launch(void* const* d_in, const int* in_sizes, int n_in,
                              void* d_out, int out_size, void* d_ws, size_t ws_size,
                              hipStream_t stream) {
    (void)in_sizes; (void)n_in; (void)out_size; (void)ws_size;
    const float* x    = (const float*)d_in[0];
    const float* W    = (const float*)d_in[1];
    const float* bias = (const float*)d_in[2];
    float* out   = (float*)d_out;
    float* feats = (float*)d_ws;                 // 4096*32*4 = 512 KB scratch

    stats_kernel<<<B_DIM * C_DIM, NTH, 0, stream>>>(x, feats);
    gemv_wmma<<<B_DIM / 16, 32, 0, stream>>>(feats, W, bias, out);
}